// VariationalGNN_50766513439402
// MI455X (gfx1250) — compile-verified
//
#include <hip/hip_runtime.h>
#include <hip/hip_bf16.h>
#include <math.h>

// Problem constants (from reference)
#define BB   64
#define NN   88
#define HH   128
#define GG   128
#define DIN  256
#define MV   (BB * NN)   // 5632 rows

typedef float v2f __attribute__((ext_vector_type(2)));
typedef float v8f __attribute__((ext_vector_type(8)));

__device__ __forceinline__ v8f wmma4(v2f a, v2f b, v8f c) {
  return __builtin_amdgcn_wmma_f32_16x16x4_f32(false, a, false, b, (short)0, c,
                                               false, false);
}

// ---------------------------------------------------------------------------
// WMMA fp32 GEMM:  C = act(A1 @ W1 [+ A2 @ W2] + bias)
// A: [M x K] row-major; W: [K x N] row-major; C: [M x N] row-major.
// K, N compile-time so all load offsets fold into immediate fields.
// Each wave computes a 32x32 C tile (4 accumulators): per k-step
// 2 A-frag b64 loads + 4 B b32 loads feed 4 V_WMMA_F32_16X16X4_F32.
// Block = (32,4): 4 waves stacked along M -> 128 rows per block.
// grid = (M/128, N/32).  ACT: 0 = identity, 1 = relu, 2 = tanh
// ---------------------------------------------------------------------------
template <int ACT, bool DUAL, int K, int N>
__global__ __launch_bounds__(128) void gemm_wmma_f32(
    const float* __restrict__ A1, const float* __restrict__ W1,
    const float* __restrict__ A2, const float* __restrict__ W2,
    const float* __restrict__ bias, float* __restrict__ C) {
  const int lane  = threadIdx.x;            // 0..31 (wave32)
  const int half  = lane >> 4;              // 0 or 1
  const int lq    = lane & 15;              // 0..15
  const int tileM = (blockIdx.x * 4 + threadIdx.y) * 32;
  const int tileN = blockIdx.y * 32;
  const int row0  = tileM + lq;
  const int row1  = tileM + 16 + lq;
  const int col0  = tileN + lq;
  const int col1  = tileN + 16 + lq;
  const int koff  = half * 2;               // K sub-offset per ISA layout

  v8f acc00 = {}, acc01 = {}, acc10 = {}, acc11 = {};

  {
    const float* ap0 = A1 + (size_t)row0 * K + koff;
    const float* ap1 = A1 + (size_t)row1 * K + koff;
    const float* wp0 = W1 + koff * N + col0;
    const float* wp1 = W1 + koff * N + col1;
#pragma unroll
    for (int k = 0; k < K; k += 4) {
      v2f a0 = *(const v2f*)(ap0 + k);      // A[row0][k+koff .. +1]
      v2f a1 = *(const v2f*)(ap1 + k);
      v2f b0; b0.x = wp0[k * N]; b0.y = wp0[k * N + N];
      v2f b1; b1.x = wp1[k * N]; b1.y = wp1[k * N + N];
      acc00 = wmma4(a0, b0, acc00);
      acc01 = wmma4(a0, b1, acc01);
      acc10 = wmma4(a1, b0, acc10);
      acc11 = wmma4(a1, b1, acc11);
    }
  }
  if (DUAL) {
    const float* ap0 = A2 + (size_t)row0 * K + koff;
    const float* ap1 = A2 + (size_t)row1 * K + koff;
    const float* wp0 = W2 + koff * N + col0;
    const float* wp1 = W2 + koff * N + col1;
#pragma unroll
    for (int k = 0; k < K; k += 4) {
      v2f a0 = *(const v2f*)(ap0 + k);
      v2f a1 = *(const v2f*)(ap1 + k);
      v2f b0; b0.x = wp0[k * N]; b0.y = wp0[k * N + N];
      v2f b1; b1.x = wp1[k * N]; b1.y = wp1[k * N + N];
      acc00 = wmma4(a0, b0, acc00);
      acc01 = wmma4(a0, b1, acc01);
      acc10 = wmma4(a1, b0, acc10);
      acc11 = wmma4(a1, b1, acc11);
    }
  }

  const float bb0 = bias[col0];
  const float bb1 = bias[col1];
  // C/D layout: VGPR r -> row (base + r + 8*half), col = base + lq
#pragma unroll
  for (int r = 0; r < 8; ++r) {
    const int r0 = tileM + half * 8 + r;
    const int r1 = r0 + 16;
    float v00 = acc00[r] + bb0, v01 = acc01[r] + bb1;
    float v10 = acc10[r] + bb0, v11 = acc11[r] + bb1;
    if (ACT == 1) {
      v00 = fmaxf(v00, 0.0f); v01 = fmaxf(v01, 0.0f);
      v10 = fmaxf(v10, 0.0f); v11 = fmaxf(v11, 0.0f);
    }
    if (ACT == 2) {
      v00 = tanhf(v00); v01 = tanhf(v01);
      v10 = tanhf(v10); v11 = tanhf(v11);
    }
    C[(size_t)r0 * N + col0] = v00;
    C[(size_t)r0 * N + col1] = v01;
    C[(size_t)r1 * N + col0] = v10;
    C[(size_t)r1 * N + col1] = v11;
  }
}

// ---------------------------------------------------------------------------
// Per-graph "max over all other nodes" (exact segment_max over the
// fully-connected edge set, via top-2 per feature).
// grid = B graphs, blockDim = D features (128 or 256).
// ---------------------------------------------------------------------------
__global__ void segmax_excl_kernel(const float* __restrict__ m,
                                   float* __restrict__ out, int D) {
  const int g = blockIdx.x;
  const int f = threadIdx.x;
  const float* base = m + (size_t)g * NN * D + f;
  float max1 = -INFINITY, max2 = -INFINITY;
  int arg = -1;
  for (int v = 0; v < NN; ++v) {
    float x = base[(size_t)v * D];
    if (x > max1) { max2 = max1; max1 = x; arg = v; }
    else if (x > max2) { max2 = x; }
  }
  float* ob = out + (size_t)g * NN * D + f;
  for (int v = 0; v < NN; ++v) ob[(size_t)v * D] = (v == arg) ? max2 : max1;
}

// ---------------------------------------------------------------------------
// Readout: mean over 88 nodes + concat(hideout_obs[2], timestep_obs[1])
// grid = B, blockDim = 131
// ---------------------------------------------------------------------------
__global__ void readout_kernel(const float* __restrict__ x,
                               const float* __restrict__ hideout,
                               const float* __restrict__ ts,
                               float* __restrict__ out) {
  const int g = blockIdx.x;
  const int c = threadIdx.x;
  const int OC = GG + 3;  // 131
  if (c < GG) {
    const float* p = x + (size_t)g * NN * GG + c;
    float s = 0.0f;
    for (int v = 0; v < NN; ++v) s += p[(size_t)v * GG];
    out[(size_t)g * OC + c] = s * (1.0f / (float)NN);
  } else if (c < GG + 2) {
    out[(size_t)g * OC + c] = hideout[g * 2 + (c - GG)];
  } else if (c == GG + 2) {
    out[(size_t)g * OC + c] = ts[g];
  }
}

// ---------------------------------------------------------------------------
extern "C" void kernel_launch(void* const* d_in, const int* in_sizes, int n_in,
                              void* d_out, int out_size, void* d_ws, size_t ws_size,
                              hipStream_t stream) {
  const float* hn      = (const float*)d_in[0];   // [5632, 256]
  const float* hideout = (const float*)d_in[1];   // [64, 2]
  const float* tsobs   = (const float*)d_in[2];   // [64, 1]
  const float* Wp1     = (const float*)d_in[3];   // [256, 256]
  const float* bp1     = (const float*)d_in[4];   // [256]
  const float* Ws1     = (const float*)d_in[5];   // [256, 128]
  const float* Wn1     = (const float*)d_in[6];   // [256, 128]
  const float* b1      = (const float*)d_in[7];   // [128]
  const float* Wp2     = (const float*)d_in[8];   // [128, 128]
  const float* bp2     = (const float*)d_in[9];   // [128]
  const float* Ws2     = (const float*)d_in[10];  // [128, 128]
  const float* Wn2     = (const float*)d_in[11];  // [128, 128]
  const float* b2      = (const float*)d_in[12];  // [128]
  // d_in[13]/d_in[14] = src/dst edge lists: unused — graph is fully connected,
  // segment_max collapses to per-graph exclusive max (top-2 trick).

  float* ws = (float*)d_ws;
  float* m1    = ws;                         // [5632*256]  (reused for m2, out2)
  float* ngh   = ws + (size_t)MV * DIN;      // [5632*256]  (neigh1, later neigh2)
  float* res1  = ngh + (size_t)MV * DIN;     // [5632*128]
  float* m2    = m1;                         // [5632*128]  overwrites m1 (done)
  float* ngh2  = ngh;                        // [5632*128]  overwrites neigh1 (done)
  float* out2  = m1;                         // [5632*128]  overwrites m2 (done)

  dim3 blk(32, 4);
  const int gx = MV / 128;                   // 44 blocks along M

  // Layer 1: m1 = relu(hn @ Wp1 + bp1)            M=5632 K=256 N=256
  gemm_wmma_f32<1, false, DIN, DIN><<<dim3(gx, DIN / 32), blk, 0, stream>>>(
      hn, Wp1, nullptr, nullptr, bp1, m1);
  // neigh1 = exclusive max per graph
  segmax_excl_kernel<<<BB, DIN, 0, stream>>>(m1, ngh, DIN);
  // res1 = tanh(hn @ Ws1 + neigh1 @ Wn1 + b1)     M=5632 K=256 N=128
  gemm_wmma_f32<2, true, DIN, HH><<<dim3(gx, HH / 32), blk, 0, stream>>>(
      hn, Ws1, ngh, Wn1, b1, res1);

  // Layer 2: m2 = relu(res1 @ Wp2 + bp2)          M=5632 K=128 N=128
  gemm_wmma_f32<1, false, HH, HH><<<dim3(gx, HH / 32), blk, 0, stream>>>(
      res1, Wp2, nullptr, nullptr, bp2, m2);
  segmax_excl_kernel<<<BB, HH, 0, stream>>>(m2, ngh2, HH);
  // out2 = res1 @ Ws2 + neigh2 @ Wn2 + b2         M=5632 K=128 N=128
  gemm_wmma_f32<0, true, HH, GG><<<dim3(gx, GG / 32), blk, 0, stream>>>(
      res1, Ws2, ngh2, Wn2, b2, out2);

  // Readout: mean over nodes + concat extras -> [64, 131]
  readout_kernel<<<BB, GG + 3, 0, stream>>>(out2, hideout, tsobs, (float*)d_out);
}